// SpikingSHDNet_MLP_50732153700701
// MI455X (gfx1250) — compile-verified
//
#include <hip/hip_runtime.h>
#include <math.h>

typedef float v2f __attribute__((ext_vector_type(2)));
typedef float v8f __attribute__((ext_vector_type(8)));
typedef int   v4i __attribute__((vector_size(16)));   // matches builtin's V4i

#define TT 16
#define NB 4096
#define HH 128
#define CIN 700
#define NC 20

#if __has_builtin(__builtin_amdgcn_global_load_async_to_lds_b128) && \
    __has_builtin(__builtin_amdgcn_s_wait_asynccnt)
#define USE_ASYNC_LDS 1
typedef __attribute__((address_space(1))) v4i* g4ptr;
typedef __attribute__((address_space(3))) v4i* l4ptr;
__device__ __forceinline__ void async_copy16(const float* g, float* l) {
    __builtin_amdgcn_global_load_async_to_lds_b128(
        (g4ptr)(void*)g, (l4ptr)(void*)l, 0, 0);
}
#else
#define USE_ASYNC_LDS 0
#endif

// ---------------------------------------------------------------------------
// fp32 GEMM via V_WMMA_F32_16X16X4_F32 with double-buffered async staging.
//   C[M,Ncols] = A[M,K] @ B[K,Ncols] + bias[Ncols]
// Block = 256 threads (8 wave32s). Block tile = 128(M) x NT*16(N).
// Each wave owns one 16-row M stripe and NT accumulators; the A fragment is
// reused across all NT column tiles, so A is read from HBM exactly once.
// K-chunks (KC=32) are staged global->LDS with GLOBAL_LOAD_ASYNC_TO_LDS_B128
// (ASYNCcnt), two LDS buffers deep: chunk i+1's copies are issued before
// waiting asynccnt<=NASYNC for chunk i, overlapping HBM latency with the
// 8*NT WMMAs of the current chunk. Partial (tail) chunks issue a variable
// number of async ops, so the pipeline conservatively waits 0 around them.
// APAD=36: rows 16B-aligned for B128 and bank-conflict-free
// (r*36 mod 64 distinct for r=0..15).
// ---------------------------------------------------------------------------
template <int NT>
__global__ __launch_bounds__(256)
void gemm_wmma_f32(const float* __restrict__ A, int lda,
                   const float* __restrict__ B, int ldb,
                   const float* __restrict__ bias,
                   float* __restrict__ C, int ldc,
                   int M, int K, int Ncols)
{
    constexpr int KC     = 32;
    constexpr int APAD   = 36;
    constexpr int BW     = NT * 16 + 4;   // 16B-aligned row stride
    constexpr int NASYNC = 4 + NT / 2;    // async ops / thread / full chunk
    __shared__ float As[2][128 * APAD];
    __shared__ float Bs[2][KC * BW];

    const int tid  = threadIdx.x;
    const int wave = tid >> 5;
    const int lane = tid & 31;
    const int m0   = blockIdx.y * 128;
    const int n0   = blockIdx.x * (NT * 16);

    // fp32 WMMA lane mapping (ISA 7.12.2):
    //  A 16x4: lanes 0-15 rows M=0..15 with K={0,1}; lanes 16-31 K={2,3}
    //  B 4x16: VGPR0 = rows K={0|2}, VGPR1 = rows K={1|3}, cols across lanes
    const int rowA = lane & 15;
    const int kb   = (lane >> 4) * 2;
    const int col  = lane & 15;

    v8f acc[NT];
    #pragma unroll
    for (int nt = 0; nt < NT; ++nt) acc[nt] = v8f{};

    auto stage = [&](int k0, int sb) {
        // ---- A: 128 rows x 32 K, float4 granularity (4 ops/thread) ----
        for (int e = tid; e < 128 * 8; e += 256) {
            int r = e >> 3, c4 = (e & 7) * 4;
            int gm = m0 + r, gk = k0 + c4;
            float* dst = &As[sb][r * APAD + c4];
#if USE_ASYNC_LDS
            if (gm < M && gk + 3 < K) {
                async_copy16(A + (size_t)gm * lda + gk, dst);
            } else
#endif
            {
                #pragma unroll
                for (int j = 0; j < 4; ++j)
                    dst[j] = (gm < M && gk + j < K) ? A[(size_t)gm * lda + gk + j] : 0.0f;
            }
        }
        // ---- B: 32 K x NT*16 cols, float4 granularity (NT/2 ops/thread) ----
        for (int e = tid; e < KC * NT * 4; e += 256) {
            int r = e / (NT * 4), c4 = (e % (NT * 4)) * 4;
            int gk = k0 + r, gn = n0 + c4;
            float* dst = &Bs[sb][r * BW + c4];
#if USE_ASYNC_LDS
            if (gk < K && gn + 3 < Ncols) {
                async_copy16(B + (size_t)gk * ldb + gn, dst);
            } else
#endif
            {
                #pragma unroll
                for (int j = 0; j < 4; ++j)
                    dst[j] = (gk < K && gn + j < Ncols) ? B[(size_t)gk * ldb + gn + j] : 0.0f;
            }
        }
    };

    const int nch = (K + KC - 1) / KC;
    stage(0, 0);
    for (int ci = 0; ci < nch; ++ci) {
        const int sb = ci & 1;
        const bool have_next = (ci + 1 < nch);
        if (have_next) stage((ci + 1) * KC, sb ^ 1);
#if USE_ASYNC_LDS
        // Partial wait is only safe if the newer in-flight chunk issued
        // exactly NASYNC async ops per thread (no K/M/N tails).
        const bool next_full = have_next && ((ci + 2) * KC <= K) &&
                               (m0 + 128 <= M) && (n0 + NT * 16 <= Ncols);
        if (next_full) __builtin_amdgcn_s_wait_asynccnt(NASYNC);
        else           __builtin_amdgcn_s_wait_asynccnt(0);
#endif
        __syncthreads();

        const int arow = wave * 16 + rowA;
        #pragma unroll
        for (int kk = 0; kk < KC; kk += 4) {
            v2f a;
            a.x = As[sb][arow * APAD + kk + kb];
            a.y = As[sb][arow * APAD + kk + kb + 1];
            #pragma unroll
            for (int nt = 0; nt < NT; ++nt) {
                v2f b;
                b.x = Bs[sb][(kk + kb) * BW + nt * 16 + col];
                b.y = Bs[sb][(kk + kb + 1) * BW + nt * 16 + col];
                acc[nt] = __builtin_amdgcn_wmma_f32_16x16x4_f32(
                              false, a, false, b, (short)0, acc[nt], false, false);
            }
        }
        __syncthreads();
    }

    // epilogue: VGPR r holds M=r (lanes 0-15) / M=r+8 (lanes 16-31)
    const int mhalf = (lane >> 4) * 8;
    #pragma unroll
    for (int nt = 0; nt < NT; ++nt) {
        int gn = n0 + nt * 16 + col;
        float bv = (gn < Ncols) ? bias[gn] : 0.0f;
        #pragma unroll
        for (int r = 0; r < 8; ++r) {
            int gm = m0 + wave * 16 + mhalf + r;
            if (gm < M && gn < Ncols)
                C[(size_t)gm * ldc + gn] = acc[nt][r] + bv;
        }
    }
}

// ---------------------------------------------------------------------------
// LIF chain 1: a1 is [n*T+t, H] (matches x@W1 row order). Writes s1 in
// [t, n, h] layout (t-major, matching the features/attended output layout).
// ---------------------------------------------------------------------------
__global__ __launch_bounds__(256)
void lif1_kernel(const float* __restrict__ a1, float* __restrict__ s1)
{
    int idx = blockIdx.x * blockDim.x + threadIdx.x;   // n*128 + h
    if (idx >= NB * HH) return;
    int n = idx >> 7, h = idx & 127;
    float v = 0.0f;
    #pragma unroll
    for (int t = 0; t < TT; ++t) {
        float inp = a1[(size_t)(n * TT + t) * HH + h];
        v = 0.5f * (v + inp);                 // v += (inp - v)/tau, tau=2
        float s = (v - 1.0f >= 0.0f) ? 1.0f : 0.0f;
        v *= (1.0f - s);
        s1[(size_t)(t * NB + n) * HH + h] = s;
    }
}

// ---------------------------------------------------------------------------
// LIF chain 2 + fused TATCA descriptors. One block (128 threads) per sample n.
// Produces: features[t,n,h] (model output), fmean[n,h] = mean_t features,
// t_attn[n,t] = sigmoid(conv1d_same(mean_h features, wt)).
// ---------------------------------------------------------------------------
__global__ __launch_bounds__(128)
void lif2_kernel(const float* __restrict__ b2pre,
                 const float* __restrict__ wt,
                 float* __restrict__ features,
                 float* __restrict__ fmean,
                 float* __restrict__ t_attn)
{
    __shared__ float red[128];
    __shared__ float tm[TT];
    int n = blockIdx.x;
    int h = threadIdx.x;

    float v = 0.0f, fsum = 0.0f;
    for (int t = 0; t < TT; ++t) {
        float inp = b2pre[(size_t)(t * NB + n) * HH + h];
        v = 0.5f * (v + inp);
        float s = (v - 1.0f >= 0.0f) ? 1.0f : 0.0f;
        v *= (1.0f - s);
        features[(size_t)(t * NB + n) * HH + h] = s;
        fsum += s;
        // mean over h for temporal descriptor
        red[h] = s;
        __syncthreads();
        #pragma unroll
        for (int off = 64; off > 0; off >>= 1) {
            if (h < off) red[h] += red[h + off];
            __syncthreads();
        }
        if (h == 0) tm[t] = red[0] * (1.0f / 128.0f);
        __syncthreads();
    }
    fmean[(size_t)n * HH + h] = fsum * (1.0f / (float)TT);

    if (h < TT) {  // cross-correlation, kernel 3, SAME zero padding
        float w0 = wt[0], w1 = wt[1], w2 = wt[2];
        float acc = 0.0f;
        if (h - 1 >= 0) acc += tm[h - 1] * w0;
        acc += tm[h] * w1;
        if (h + 1 < TT) acc += tm[h + 1] * w2;
        t_attn[n * TT + h] = 1.0f / (1.0f + __expf(-acc));
    }
}

// ---------------------------------------------------------------------------
// attended[t,n,h] = features[t,n,h] * t_attn[n,t] * sigmoid(cpre[n,h])
// ---------------------------------------------------------------------------
__global__ __launch_bounds__(256)
void attend_kernel(const float* __restrict__ features,
                   const float* __restrict__ t_attn,
                   const float* __restrict__ cpre,
                   float* __restrict__ attended)
{
    size_t idx = (size_t)blockIdx.x * blockDim.x + threadIdx.x;
    if (idx >= (size_t)TT * NB * HH) return;
    int t = (int)(idx / ((size_t)NB * HH));
    int r = (int)(idx % ((size_t)NB * HH));
    int n = r >> 7, h = r & 127;
    float ca = 1.0f / (1.0f + __expf(-cpre[(size_t)n * HH + h]));
    attended[idx] = features[idx] * t_attn[n * TT + t] * ca;
}

// ---------------------------------------------------------------------------
// LIF chain 3 with the T-mean folded before the final W4 GEMM:
// ss[n,h] = mean_t s3[t,n,h]
// ---------------------------------------------------------------------------
__global__ __launch_bounds__(256)
void lif3_kernel(const float* __restrict__ a3, float* __restrict__ ss)
{
    int idx = blockIdx.x * blockDim.x + threadIdx.x;
    if (idx >= NB * HH) return;
    int n = idx >> 7, h = idx & 127;
    float v = 0.0f, ssum = 0.0f;
    #pragma unroll
    for (int t = 0; t < TT; ++t) {
        float inp = a3[(size_t)(t * NB + n) * HH + h];
        v = 0.5f * (v + inp);
        float s = (v - 1.0f >= 0.0f) ? 1.0f : 0.0f;
        v *= (1.0f - s);
        ssum += s;
    }
    ss[(size_t)n * HH + h] = ssum * (1.0f / (float)TT);
}

// ---------------------------------------------------------------------------
extern "C" void kernel_launch(void* const* d_in, const int* in_sizes, int n_in,
                              void* d_out, int out_size, void* d_ws, size_t ws_size,
                              hipStream_t stream) {
    const float* x  = (const float*)d_in[0];
    const float* W1 = (const float*)d_in[1];
    const float* b1 = (const float*)d_in[2];
    const float* W2 = (const float*)d_in[3];
    const float* b2 = (const float*)d_in[4];
    const float* W3 = (const float*)d_in[5];
    const float* b3 = (const float*)d_in[6];
    const float* W4 = (const float*)d_in[7];
    const float* b4 = (const float*)d_in[8];
    const float* wt = (const float*)d_in[9];
    const float* Wc = (const float*)d_in[10];
    const float* bc = (const float*)d_in[11];

    float* out      = (float*)d_out;
    float* features = out + (size_t)NB * NC;            // [T,N,H]
    float* attended = features + (size_t)TT * NB * HH;  // [T,N,H]

    float* ws    = (float*)d_ws;
    const size_t BIG = (size_t)TT * NB * HH;            // 8,388,608 floats
    float* buf0  = ws;            // A1 / b2pre / cpre / a3
    float* buf1  = ws + BIG;      // s1 / fmean / ss
    float* tattn = ws + 2 * BIG;  // [N,T]

    const int M_big = NB * TT;    // 65536

    // 1) A1 = x @ W1 + b1   (x viewed as [N*T, CIN], row m = n*T + t)
    gemm_wmma_f32<8><<<dim3(1, M_big / 128), 256, 0, stream>>>(
        x, CIN, W1, HH, b1, buf0, HH, M_big, CIN, HH);

    // 2) v1 chain -> s1 [t,n,h]
    lif1_kernel<<<(NB * HH + 255) / 256, 256, 0, stream>>>(buf0, buf1);

    // 3) b2pre = s1 @ W2 + b2
    gemm_wmma_f32<8><<<dim3(1, M_big / 128), 256, 0, stream>>>(
        buf1, HH, W2, HH, b2, buf0, HH, M_big, HH, HH);

    // 4) v2 chain -> features (output) + fmean + t_attn
    lif2_kernel<<<NB, 128, 0, stream>>>(buf0, wt, features, buf1, tattn);

    // 5) cpre = fmean @ Wc + bc
    gemm_wmma_f32<8><<<dim3(1, NB / 128), 256, 0, stream>>>(
        buf1, HH, Wc, HH, bc, buf0, HH, NB, HH, HH);

    // 6) attended = features * t_attn * sigmoid(cpre)   (output)
    attend_kernel<<<(int)((BIG + 255) / 256), 256, 0, stream>>>(
        features, tattn, buf0, attended);

    // 7) a3 = attended @ W3 + b3
    gemm_wmma_f32<8><<<dim3(1, M_big / 128), 256, 0, stream>>>(
        attended, HH, W3, HH, b3, buf0, HH, M_big, HH, HH);

    // 8) v3 chain, T-mean folded -> ss[n,h]
    lif3_kernel<<<(NB * HH + 255) / 256, 256, 0, stream>>>(buf0, buf1);

    // 9) out = ss @ W4 + b4   [4096,128] x [128,20]
    gemm_wmma_f32<2><<<dim3(1, NB / 128), 256, 0, stream>>>(
        buf1, HH, W4, NC, b4, out, NC, NB, HH, NC);
}